// NaiveGCNConv_3307124818739
// MI455X (gfx1250) — compile-verified
//
#include <hip/hip_runtime.h>
#include <hip/hip_bf16.h>

typedef __attribute__((ext_vector_type(2))) float v2f;
typedef __attribute__((ext_vector_type(8))) float v8f;

#define NUM_NODES 50000
#define NUM_EDGES 1600000
#define IN_C 512
#define OUT_C 512
#define NUM_STRIPS (NUM_NODES / 16)   // 3125, exact

// ---------------- atomic helper: relaxed agent-scope f32 add -> global_atomic_add_f32
__device__ __forceinline__ void atomic_add_f32(float* p, float v) {
    __hip_atomic_fetch_add(p, v, __ATOMIC_RELAXED, __HIP_MEMORY_SCOPE_AGENT);
}

// ---------------- degree init: deg[n] = 1.0 (self loop)
__global__ void gcn_deg_init(float* __restrict__ deg) {
    int n = blockIdx.x * blockDim.x + threadIdx.x;
    if (n < NUM_NODES) deg[n] = 1.0f;
}

// ---------------- degree count: one thread per edge, scatter-add over src
__global__ void gcn_deg_count(const int* __restrict__ src, float* __restrict__ deg) {
    int e = blockIdx.x * blockDim.x + threadIdx.x;
    if (e < NUM_EDGES) atomic_add_f32(&deg[src[e]], 1.0f);
}

// ---------------- dis = deg^-1/2, dinv = deg^-1
__global__ void gcn_deg_final(const float* __restrict__ deg,
                              float* __restrict__ dis,
                              float* __restrict__ dinv) {
    int n = blockIdx.x * blockDim.x + threadIdx.x;
    if (n < NUM_NODES) {
        float d = deg[n];
        dis[n]  = __frsqrt_rn(d);
        dinv[n] = 1.0f / d;
    }
}

// ---------------- GEMM: feat = x @ W^T + b  via V_WMMA_F32_16X16X4_F32
// 2-D grid: blockIdx.x -> group of 8 row strips (one per wave, unique A data),
//           blockIdx.y -> 64-col group shared by all waves (B = W rows, L0/L2 resident).
// The 102 MB x matrix is read once per 64-col group (8 times); the 1 MB W is re-read per block.
// A fragment (16x4 f32): lanes 0-15 -> K = k0+{0,1}, lanes 16-31 -> K = k0+{2,3}.
__global__ void __launch_bounds__(256)
gcn_gemm_wmma(const float* __restrict__ x, const float* __restrict__ W,
              const float* __restrict__ b, float* __restrict__ feat) {
    const int lane  = threadIdx.x & 31;
    const int wave  = threadIdx.x >> 5;
    int strip = blockIdx.x * 8 + wave;
    if (strip >= NUM_STRIPS) strip = NUM_STRIPS - 1;  // tail waves duplicate strip 3124
                                                      // (identical recompute -> deterministic)
    const int row0  = strip * 16;
    const int n0    = blockIdx.y * 64;
    const int m     = lane & 15;
    const int khalf = lane >> 4;                      // 0 for lanes 0-15, 1 for 16-31

    const float* xrow = x + (size_t)(row0 + m) * IN_C;

    v8f acc0 = {}, acc1 = {}, acc2 = {}, acc3 = {};

    const float* wr0 = W + (size_t)(n0 + 0  + m) * IN_C;
    const float* wr1 = W + (size_t)(n0 + 16 + m) * IN_C;
    const float* wr2 = W + (size_t)(n0 + 32 + m) * IN_C;
    const float* wr3 = W + (size_t)(n0 + 48 + m) * IN_C;

    #pragma unroll 4
    for (int k0 = 0; k0 < IN_C; k0 += 4) {
        const int k = k0 + khalf * 2;                 // contiguous pair per lane
        v2f a  = *(const v2f*)(xrow + k);             // A: x[m][k..k+1]
        v2f b0 = *(const v2f*)(wr0 + k);              // B: W[n][k..k+1] == (W^T)[k][n]
        v2f b1 = *(const v2f*)(wr1 + k);
        v2f b2 = *(const v2f*)(wr2 + k);
        v2f b3 = *(const v2f*)(wr3 + k);
        acc0 = __builtin_amdgcn_wmma_f32_16x16x4_f32(false, a, false, b0, (short)0, acc0, false, false);
        acc1 = __builtin_amdgcn_wmma_f32_16x16x4_f32(false, a, false, b1, (short)0, acc1, false, false);
        acc2 = __builtin_amdgcn_wmma_f32_16x16x4_f32(false, a, false, b2, (short)0, acc2, false, false);
        acc3 = __builtin_amdgcn_wmma_f32_16x16x4_f32(false, a, false, b3, (short)0, acc3, false, false);
    }

    // C/D layout: VGPR r, lane l -> m = r + 8*(l>=16), n = n0 + 16*j + (l&15)
    const int mbase = row0 + khalf * 8;
    v8f accs[4] = {acc0, acc1, acc2, acc3};
    #pragma unroll
    for (int j = 0; j < 4; ++j) {
        const int n  = n0 + j * 16 + m;
        const float bv = b[n];
        #pragma unroll
        for (int r = 0; r < 8; ++r) {
            feat[(size_t)(mbase + r) * OUT_C + n] = accs[j][r] + bv;
        }
    }
}

// ---------------- out init with self term: out[n][c] = feat[n][c] * dinv[n]
__global__ void gcn_self_init(const float* __restrict__ feat,
                              const float* __restrict__ dinv,
                              float* __restrict__ out) {
    size_t idx = (size_t)blockIdx.x * blockDim.x + threadIdx.x;   // float4 index
    if (idx >= (size_t)NUM_NODES * OUT_C / 4) return;
    int n = (int)(idx >> 7);                                      // 512/4 = 128 vec4 per row
    float4 v = ((const float4*)feat)[idx];
    float s = dinv[n];
    v.x *= s; v.y *= s; v.z *= s; v.w *= s;
    ((float4*)out)[idx] = v;
}

// ---------------- edge scatter: one wave32 per edge
// out[s][:] += dis[s]*dis[t] * feat[t][:]   (512 f32 per edge, float4 gathers,
// feat is L2-resident at 102 MB / 192 MB L2; atomics resolve at L2)
__global__ void __launch_bounds__(256)
gcn_scatter(const int* __restrict__ src, const int* __restrict__ tar,
            const float* __restrict__ dis, const float* __restrict__ feat,
            float* __restrict__ out) {
    size_t gid = (size_t)blockIdx.x * blockDim.x + threadIdx.x;
    int e    = (int)(gid >> 5);
    int lane = (int)(gid & 31);
    if (e >= NUM_EDGES) return;

    int s = src[e];
    int t = tar[e];
    float w = dis[s] * dis[t];

    const float4* fp = (const float4*)(feat + (size_t)t * OUT_C);
    float*        op = out + (size_t)s * OUT_C;

    #pragma unroll
    for (int j = 0; j < 4; ++j) {
        int vidx = lane + j * 32;            // 128 float4 = 512 floats
        float4 v = fp[vidx];
        int c = vidx * 4;
        atomic_add_f32(op + c + 0, w * v.x);
        atomic_add_f32(op + c + 1, w * v.y);
        atomic_add_f32(op + c + 2, w * v.z);
        atomic_add_f32(op + c + 3, w * v.w);
    }
}

extern "C" void kernel_launch(void* const* d_in, const int* in_sizes, int n_in,
                              void* d_out, int out_size, void* d_ws, size_t ws_size,
                              hipStream_t stream) {
    const float* x  = (const float*)d_in[0];
    const int* eidx = (const int*)d_in[1];        // (2, E): src then tar
    const float* W  = (const float*)d_in[2];
    const float* b  = (const float*)d_in[3];
    float* out      = (float*)d_out;

    const int* src = eidx;
    const int* tar = eidx + NUM_EDGES;

    // workspace layout
    char* ws = (char*)d_ws;
    float* feat = (float*)ws;                                    // 50000*512*4 = 102.4 MB
    float* deg  = (float*)(ws + (size_t)NUM_NODES * OUT_C * 4);  // +200 KB
    float* dis  = deg  + NUM_NODES;
    float* dinv = dis  + NUM_NODES;

    // 1) degree (self-loop init + scatter count + normalize)
    gcn_deg_init <<<(NUM_NODES + 255) / 256, 256, 0, stream>>>(deg);
    gcn_deg_count<<<(NUM_EDGES + 255) / 256, 256, 0, stream>>>(src, deg);
    gcn_deg_final<<<(NUM_NODES + 255) / 256, 256, 0, stream>>>(deg, dis, dinv);

    // 2) feat = x @ W^T + b   (f32 WMMA; waves own distinct row strips, cols shared)
    {
        dim3 grid((NUM_STRIPS + 7) / 8, OUT_C / 64);   // 391 x 8
        gcn_gemm_wmma<<<grid, 256, 0, stream>>>(x, W, b, feat);
    }

    // 3) out = feat * deg^-1  (self contribution, also initializes poisoned d_out)
    {
        size_t nvec4 = (size_t)NUM_NODES * OUT_C / 4;
        gcn_self_init<<<(unsigned)((nvec4 + 255) / 256), 256, 0, stream>>>(feat, dinv, out);
    }

    // 4) out[src] += w * feat[tar]  (one wave per edge, fp32 L2 atomics)
    {
        size_t threads = (size_t)NUM_EDGES * 32;
        gcn_scatter<<<(unsigned)((threads + 255) / 256), 256, 0, stream>>>(src, tar, dis, feat, out);
    }
}